// GraphSage_19215683683123
// MI455X (gfx1250) — compile-verified
//
#include <hip/hip_runtime.h>

// GraphSAGE 2-hop mean aggregation for MI455X (gfx1250).
// Memory-bound (AI ~0.2 FLOP/B, working set L2-resident). Strategy:
//  - gather rows via CDNA5 async global->LDS (ASYNCcnt), 4-deep pipelined per wave
//  - exploit sorted edge_dst: accumulate runs in registers, flush with f32 atomics
//  - output layout: [0:64)=features, [64:128)=agg1, [128:256)=agg2 ; h1 = out[:,0:128)

#define WPB  4     // waves per block (wave32)
#define EPW  128   // edges per wave
#define PF   3     // async prefetch distance (edges ahead)
#define NBUF 4     // LDS row buffers per wave (power of 2, > PF)

#if __has_builtin(__builtin_amdgcn_global_load_async_to_lds_b128) && \
    __has_builtin(__builtin_amdgcn_global_load_async_to_lds_b64)  && \
    __has_builtin(__builtin_amdgcn_s_wait_asynccnt)
#define USE_ASYNC 1
#warning "CDNA5 async-to-LDS builtins: AVAILABLE (async path compiled)"
#else
#define USE_ASYNC 0
#warning "CDNA5 async-to-LDS builtins: NOT available (direct-load fallback compiled)"
#endif

typedef int v2i __attribute__((ext_vector_type(2)));
typedef int v4i __attribute__((ext_vector_type(4)));
typedef __attribute__((address_space(1))) v2i* g_v2i_p;
typedef __attribute__((address_space(3))) v2i* l_v2i_p;
typedef __attribute__((address_space(1))) v4i* g_v4i_p;
typedef __attribute__((address_space(3))) v4i* l_v4i_p;

template<int VW>
__device__ __forceinline__ void async_row_to_lds(const float* g, float* l) {
#if USE_ASYNC
  if constexpr (VW == 4) {
    __builtin_amdgcn_global_load_async_to_lds_b128((g_v4i_p)(g), (l_v4i_p)(l), 0, 0);
  } else {
    __builtin_amdgcn_global_load_async_to_lds_b64((g_v2i_p)(g), (l_v2i_p)(l), 0, 0);
  }
#else
  (void)g; (void)l;
#endif
}

// Segmented gather/scatter-add body. Each wave owns EPW contiguous edges; lane owns
// VW consecutive feature columns (32*VW floats per source row). Source rows are staged
// through a NBUF-deep LDS ring by the async copy engine; destination runs (edge_dst is
// sorted) are accumulated in registers and flushed with one atomic burst per run.
template<int VW>
__device__ __forceinline__ void
scatter_impl(const float* __restrict__ SP, int srcStride,
             const int* __restrict__ esrc, const int* __restrict__ edst,
             float* __restrict__ OP, int dstColOff, int E,
             float* __restrict__ lbase /* this wave's NBUF*32*VW floats of LDS */)
{
  constexpr int ROWF = 32 * VW;
  const int lane = threadIdx.x & 31;

  const long wave  = (long)blockIdx.x * WPB + (threadIdx.x >> 5);
  const long start = wave * (long)EPW;
  if (start >= E) return;
  const int cnt = (int)(((long)E - start) < (long)EPW ? ((long)E - start) : (long)EPW);

  float acc[VW];
#pragma unroll
  for (int k = 0; k < VW; ++k) acc[k] = 0.f;

  // prime the async pipeline: edges 0..PF-1
  for (int j = 0; j < PF && j < cnt; ++j) {
    int s = __builtin_amdgcn_readfirstlane(esrc[start + j]);
    async_row_to_lds<VW>(SP + (size_t)s * srcStride + lane * VW,
                         lbase + j * ROWF + lane * VW);
  }

  int d = __builtin_amdgcn_readfirstlane(edst[start]);

  for (int i = 0; i < cnt; ++i) {
    if (i + PF < cnt) {
      int s = __builtin_amdgcn_readfirstlane(esrc[start + i + PF]);
      async_row_to_lds<VW>(SP + (size_t)s * srcStride + lane * VW,
                           lbase + ((i + PF) & (NBUF - 1)) * ROWF + lane * VW);
    }

#if USE_ASYNC
    // async copies complete in order: allow min(cnt-1-i, PF) still in flight
    const int allowed = (i + PF < cnt) ? PF : (cnt - 1 - i);
    if      (allowed <= 0) __builtin_amdgcn_s_wait_asynccnt(0);
    else if (allowed == 1) __builtin_amdgcn_s_wait_asynccnt(1);
    else if (allowed == 2) __builtin_amdgcn_s_wait_asynccnt(2);
    else                   __builtin_amdgcn_s_wait_asynccnt(3);
    const float* lb = lbase + (i & (NBUF - 1)) * ROWF + lane * VW;
#else
    int si = __builtin_amdgcn_readfirstlane(esrc[start + i]);
    const float* lb = SP + (size_t)si * srcStride + lane * VW;
#endif
#pragma unroll
    for (int k = 0; k < VW; ++k) acc[k] += lb[k];

    int dn = -1;
    if (i + 1 < cnt) dn = __builtin_amdgcn_readfirstlane(edst[start + i + 1]);

    // flush accumulated run when destination changes (or at chunk end)
    if (dn != d) {
      float* op = OP + (size_t)d * 256 + dstColOff + lane * VW;
#pragma unroll
      for (int k = 0; k < VW; ++k) { atomicAdd(op + k, acc[k]); acc[k] = 0.f; }
      d = dn;
    }
  }
}

// hop 2: gather h1 rows (= out cols [0,128), stride 256), scatter-add into cols [128,256)
__global__ void __launch_bounds__(WPB * 32)
k_scatter_h2(const float* __restrict__ SP, const int* __restrict__ esrc,
             const int* __restrict__ edst, float* __restrict__ OP, int E)
{
  __shared__ __align__(16) float smem[WPB][NBUF][128];
  scatter_impl<4>(SP, 256, esrc, edst, OP, 128, E, &smem[threadIdx.x >> 5][0][0]);
}

// hop 1: gather feature rows (64 f32, stride 64), scatter-add into cols [64,128)
__global__ void __launch_bounds__(WPB * 32)
k_scatter_h1(const float* __restrict__ SP, const int* __restrict__ esrc,
             const int* __restrict__ edst, float* __restrict__ OP, int E)
{
  __shared__ __align__(16) float smem[WPB][NBUF][64];
  scatter_impl<2>(SP, 64, esrc, edst, OP, 64, E, &smem[threadIdx.x >> 5][0][0]);
}

// out[:,0:64) = features ; out[:,64:256) = 0 ; deg = 0
__global__ void k_init(const float* __restrict__ F, float* __restrict__ O,
                       float* __restrict__ deg, int N)
{
  long i = (long)blockIdx.x * blockDim.x + threadIdx.x;
  if (i >= (long)N * 64) return;
  int n = (int)(i >> 6);
  int c = (int)(i & 63);
  float v = F[i];
  float* row = O + (size_t)n * 256;
  row[c]       = v;
  row[64 + c]  = 0.f;
  row[128 + c] = 0.f;
  row[192 + c] = 0.f;
  if (c == 0) deg[n] = 0.f;
}

// sorted edge_dst: one atomic per run instead of per edge (~16x fewer atomics)
__global__ void k_degree(const int* __restrict__ edst, float* __restrict__ deg, int E)
{
  long e = (long)blockIdx.x * blockDim.x + threadIdx.x;
  if (e >= E) return;
  int d = edst[e];
  if (e > 0 && edst[e - 1] == d) return;  // not a run start
  int len = 1;
  while (e + len < E && edst[e + len] == d) ++len;
  atomicAdd(&deg[d], (float)len);
}

// out[:, colOff : colOff+cols) /= max(deg, 1)
__global__ void k_norm(float* __restrict__ O, const float* __restrict__ deg,
                       int N, int colOff, int cols)
{
  long i = (long)blockIdx.x * blockDim.x + threadIdx.x;
  long total = (long)N * cols;
  if (i >= total) return;
  int n = (int)(i / cols);
  int c = (int)(i % cols);
  float dg = deg[n];
  dg = dg < 1.0f ? 1.0f : dg;
  O[(size_t)n * 256 + colOff + c] *= (1.0f / dg);
}

extern "C" void kernel_launch(void* const* d_in, const int* in_sizes, int n_in,
                              void* d_out, int out_size, void* d_ws, size_t ws_size,
                              hipStream_t stream)
{
  const float* F    = (const float*)d_in[0];
  const int*   esrc = (const int*)d_in[1];   // edge indices (int32 per harness convention)
  const int*   edst = (const int*)d_in[2];   // sorted
  const int    N    = in_sizes[0] / 64;
  const int    E    = in_sizes[1];

  float* O   = (float*)d_out;   // [N, 256]
  float* deg = (float*)d_ws;    // [N]

  const long waves  = ((long)E + EPW - 1) / EPW;
  const int  blocks = (int)((waves + WPB - 1) / WPB);

  {
    long total = (long)N * 64;
    k_init<<<(int)((total + 255) / 256), 256, 0, stream>>>(F, O, deg, N);
  }
  k_degree<<<(E + 255) / 256, 256, 0, stream>>>(edst, deg, E);

  k_scatter_h1<<<blocks, WPB * 32, 0, stream>>>(F, esrc, edst, O, E);
  {
    long total = (long)N * 64;
    k_norm<<<(int)((total + 255) / 256), 256, 0, stream>>>(O, deg, N, 64, 64);
  }

  k_scatter_h2<<<blocks, WPB * 32, 0, stream>>>(O, esrc, edst, O, E);
  {
    long total = (long)N * 128;
    k_norm<<<(int)((total + 511) / 512), 512, 0, stream>>>(O, deg, N, 128, 128);
  }
}